// MaxPool_2d_22308060135988
// MI455X (gfx1250) — compile-verified
//
#include <hip/hip_runtime.h>

// 2x2/stride-2 max-pool, NCHW fp32: in 32x64x224x224 -> out 32x64x112x112.
// SAME padding is zero for (224, k=2, s=2), so this is a pure windowed max.
//
// Memory-bound streaming kernel: ~411 MB read + ~103 MB write, no reuse.
// Strategy: b128 loads/stores with non-temporal hints, 4 outputs per thread,
// fully coalesced along W, exact-cover grid (no bounds checks).

typedef float v4f __attribute__((ext_vector_type(4)));

namespace {
constexpr unsigned kN  = 32;
constexpr unsigned kC  = 64;
constexpr unsigned kH  = 224;
constexpr unsigned kW  = 224;
constexpr unsigned kHO = 112;
constexpr unsigned kWO = 112;
constexpr unsigned kGroups = kWO / 4;                        // 28 output quads per row
constexpr unsigned kTotalThreads = kN * kC * kHO * kGroups;  // 6,422,528
constexpr unsigned kBlock = 256;                             // 8 waves (wave32)
constexpr unsigned kGrid  = kTotalThreads / kBlock;          // 25,088 (exact)
constexpr unsigned kInPlane  = kH * kW;    // 50,176
constexpr unsigned kOutPlane = kHO * kWO;  // 12,544
}  // namespace

__global__ __launch_bounds__(kBlock) void maxpool2x2_s2_kernel(
    const float* __restrict__ X, float* __restrict__ Y) {
  const unsigned tid  = blockIdx.x * kBlock + threadIdx.x;
  const unsigned g    = tid % kGroups;        // which quad of output columns
  const unsigned rest = tid / kGroups;
  const unsigned h    = rest % kHO;           // output row
  const unsigned nc   = rest / kHO;           // fused batch*channel plane

  // Input base: plane nc, row 2h, column 8g. Row0/row1 are 16B-aligned
  // (row stride 896 B, column stride 32 B).
  const unsigned ibase = nc * kInPlane + (h * 2u) * kW + g * 8u;
  const v4f* __restrict__ p0 = (const v4f*)(X + ibase);        // row 2h
  const v4f* __restrict__ p1 = (const v4f*)(X + ibase + kW);   // row 2h+1

  // Four independent b128 loads, streamed once -> non-temporal.
  v4f a0 = __builtin_nontemporal_load(p0);
  v4f a1 = __builtin_nontemporal_load(p0 + 1);
  v4f b0 = __builtin_nontemporal_load(p1);
  v4f b1 = __builtin_nontemporal_load(p1 + 1);

  // 2x2 window max per output column; fmax trees fuse into v_max3_f32.
  v4f o;
  o.x = fmaxf(fmaxf(a0.x, a0.y), fmaxf(b0.x, b0.y));
  o.y = fmaxf(fmaxf(a0.z, a0.w), fmaxf(b0.z, b0.w));
  o.z = fmaxf(fmaxf(a1.x, a1.y), fmaxf(b1.x, b1.y));
  o.w = fmaxf(fmaxf(a1.z, a1.w), fmaxf(b1.z, b1.w));

  // One b128 non-temporal store of 4 consecutive output columns.
  const unsigned obase = nc * kOutPlane + h * kWO + g * 4u;
  __builtin_nontemporal_store(o, (v4f*)(Y + obase));
}

extern "C" void kernel_launch(void* const* d_in, const int* in_sizes, int n_in,
                              void* d_out, int out_size, void* d_ws, size_t ws_size,
                              hipStream_t stream) {
  const float* X = (const float*)d_in[0];
  float* Y = (float*)d_out;
  maxpool2x2_s2_kernel<<<dim3(kGrid), dim3(kBlock), 0, stream>>>(X, Y);
}